// Hausdorff_Loss_65635690217853
// MI455X (gfx1250) — compile-verified
//
#include <hip/hip_runtime.h>
#include <hip/hip_bf16.h>
#include <math.h>

typedef float v2f __attribute__((ext_vector_type(2)));
typedef float v4f __attribute__((ext_vector_type(4)));
typedef float v8f __attribute__((ext_vector_type(8)));

#define FLT_BIG 3.402823466e+38f

// ---------------------------------------------------------------------------
// Kernel 1: pack each point into 4 floats {x, y, z, -0.5*||p||^2}.
// The 4th slot rides in the WMMA K=4 pad lane so the norm bias is computed
// by the matrix unit itself.
// ---------------------------------------------------------------------------
__global__ void hd_pack_kernel(const float* __restrict__ X,
                               const float* __restrict__ Y,
                               v4f* __restrict__ xpk,
                               v4f* __restrict__ ypk,
                               int nX, int nY) {
    int i = blockIdx.x * blockDim.x + threadIdx.x;
    if (i < nX) {
        float a = X[3 * i + 0], b = X[3 * i + 1], c = X[3 * i + 2];
        v4f p = {a, b, c, -0.5f * (a * a + b * b + c * c)};
        xpk[i] = p;
    }
    if (i < nY) {
        float a = Y[3 * i + 0], b = Y[3 * i + 1], c = Y[3 * i + 2];
        v4f p = {a, b, c, -0.5f * (a * a + b * b + c * c)};
        ypk[i] = p;
    }
}

// ---------------------------------------------------------------------------
// Kernel 2: for each row i of A, rmin[i] = min_j ||A_i - B_j||^2
//                                        = ||A_i||^2 - 2 * max_j dot'(i,j),
// where dot'(i,j) = A_i . B_j - 0.5*||B_j||^2 comes straight out of
// V_WMMA_F32_16X16X4_F32: A pad (K=3) = 1.0, B pad (K=3) = -0.5*||B_j||^2.
//
// A fragment (16x4 fp32, ISA 7.12.2): lanes 0-15 carry (K=0,K=1) of row
// M=lane, lanes 16-31 carry (K=2,K=3) of row M=lane-16.
// C/D layout: lane L, slot v holds M = v + 8*(L>=16), N = L&15.
// Both halves load one aligned b64 from the packed array: offset 16*idx+8*half.
// ---------------------------------------------------------------------------
__global__ __launch_bounds__(256) void hd_rowmin_kernel(
    const v4f* __restrict__ Apk, const v4f* __restrict__ Bpk,
    float* __restrict__ rmin, int nB) {

    const float* Ap = (const float*)Apk;
    const float* Bp = (const float*)Bpk;

    const int lane = threadIdx.x & 31;
    const int half = lane >> 4;    // 0: carries K=0,1   1: carries K=2,3
    const int mrow = lane & 15;
    const int wave = (blockIdx.x * blockDim.x + threadIdx.x) >> 5;
    const int i0 = wave * 16;      // first row of this wave's A tile

    // Invariant A fragment: one b64 load + one cndmask (pad -> 1.0).
    const int ia = i0 + mrow;
    v2f afrag = *(const v2f*)(Ap + 4 * ia + 2 * half);
    afrag.y = half ? 1.0f : afrag.y;

    // -0.5*||A_M||^2 bias for the 8 M-slots this lane owns in C/D.
    float abias[8];
#pragma unroll
    for (int v = 0; v < 8; ++v) abias[v] = Ap[4 * (i0 + v + 8 * half) + 3];

    float rmx[8];
#pragma unroll
    for (int v = 0; v < 8; ++v) rmx[v] = -FLT_BIG;

    // Per-lane walking pointer into packed B: 16*j bytes... (4 floats/point).
    const float* bptr = Bp + 4 * mrow + 2 * half;
    const int ntile = nB >> 4;
    for (int jt = 0; jt < ntile; ++jt) {
        v2f bfrag = *(const v2f*)(bptr);
        bptr += 64;                 // 16 points * 4 floats

        v8f dot = {};
        dot = __builtin_amdgcn_wmma_f32_16x16x4_f32(
            /*neg_a=*/false, afrag, /*neg_b=*/false, bfrag,
            /*c_mod=*/(short)0, dot, /*reuse_a=*/false, /*reuse_b=*/false);

#pragma unroll
        for (int v = 0; v < 8; ++v)
            rmx[v] = fmaxf(rmx[v], dot[v]);
    }

    // Max across the 16 lanes sharing each M (xor net stays inside the half),
    // then d2_min = -2*(bias + maxdot).  (Clamp to >=0 happens in the final
    // reduction; min/max commute with this monotone-decreasing affine map.)
#pragma unroll
    for (int v = 0; v < 8; ++v) {
        float m = rmx[v];
        m = fmaxf(m, __shfl_xor(m, 8, 32));
        m = fmaxf(m, __shfl_xor(m, 4, 32));
        m = fmaxf(m, __shfl_xor(m, 2, 32));
        m = fmaxf(m, __shfl_xor(m, 1, 32));
        if (mrow == 0)
            rmin[i0 + v + 8 * half] = -2.0f * (abias[v] + m);
    }
}

// ---------------------------------------------------------------------------
// Kernel 3: out = sqrt(max(0, max over all nX+nY per-row squared mins)).
// ---------------------------------------------------------------------------
__global__ void hd_final_max_kernel(const float* __restrict__ vals, int n,
                                    float* __restrict__ out) {
    __shared__ float s[256];
    float m = 0.0f;
    for (int i = threadIdx.x; i < n; i += blockDim.x) m = fmaxf(m, vals[i]);
    s[threadIdx.x] = m;
    __syncthreads();
    for (int stride = 128; stride > 0; stride >>= 1) {
        if ((int)threadIdx.x < stride)
            s[threadIdx.x] = fmaxf(s[threadIdx.x], s[threadIdx.x + stride]);
        __syncthreads();
    }
    if (threadIdx.x == 0) out[0] = sqrtf(fmaxf(s[0], 0.0f));
}

// ---------------------------------------------------------------------------
extern "C" void kernel_launch(void* const* d_in, const int* in_sizes, int n_in,
                              void* d_out, int out_size, void* d_ws, size_t ws_size,
                              hipStream_t stream) {
    const float* X = (const float*)d_in[0];
    const float* Y = (const float*)d_in[1];
    const int nX = in_sizes[0] / 3;  // 8192
    const int nY = in_sizes[1] / 3;  // 8192

    float* ws  = (float*)d_ws;
    v4f* xpk   = (v4f*)ws;                       // [nX] packed {x,y,z,-n2/2}
    v4f* ypk   = (v4f*)(ws + 4 * nX);            // [nY]
    float* rX  = ws + 4 * nX + 4 * nY;           // [nX] min_j d2(i,j)
    float* rY  = rX + nX;                        // [nY] min_i d2(i,j) (contig.)

    const int nmax = (nX > nY) ? nX : nY;
    hd_pack_kernel<<<(nmax + 255) / 256, 256, 0, stream>>>(X, Y, xpk, ypk, nX, nY);

    // 8 waves per 256-thread block, 16 rows per wave -> 128 rows per block.
    hd_rowmin_kernel<<<nX / 128, 256, 0, stream>>>(xpk, ypk, rX, nY);
    hd_rowmin_kernel<<<nY / 128, 256, 0, stream>>>(ypk, xpk, rY, nX);

    hd_final_max_kernel<<<1, 256, 0, stream>>>(rX, nX + nY, (float*)d_out);
}